// UnfoldBEVProjector_481036337895
// MI455X (gfx1250) — compile-verified
//
#include <hip/hip_runtime.h>
#include <hip/hip_bf16.h>
#include <cstdint>

typedef __attribute__((ext_vector_type(2))) float v2f;
typedef __attribute__((ext_vector_type(4))) float v4f;
typedef __attribute__((ext_vector_type(8))) float v8f;

#define KPTS   40000
#define IMG_C  64
#define IMG_H  32
#define IMG_W  88
#define HW     (IMG_H * IMG_W)
#define NCAM   12
#define FEATS_ELEMS 276480000L   // 2*6*40000*9*64

// ---------------------------------------------------------------------------
// Stage 1: projection via V_WMMA_F32_16X16X4_F32.
//   D(16x16) = A(16x4) x B(4x16):
//     A rows M = cam_local*4 + i  hold P[cam][i][0..3]   (P = I4 @ E)
//     B cols N = 16 points, rows K = (x, y, -1, 1)
//   Per wave: 16 points x 4 cameras per WMMA, 3 camera groups looped.
//   VGPR layouts (ISA 7.12.2, 32-bit):
//     A: lane(M) = lane&15 ; K = (lane>>4)*2 + vgpr
//     B: lane(N) = lane&15 ; K = (lane>>4)*2 + vgpr   (dual of A)
//     D: vgpr v -> M = v + 8*(lane>>4) ; N = lane&15
// Emits packed pixel coords + float mask.  EXEC all-ones at the WMMA.
// ---------------------------------------------------------------------------
__global__ __launch_bounds__(128) void bev_project_wmma(
    const float* __restrict__ bev,      // (3, 40000)
    const float* __restrict__ Imat,     // (12, 3, 3)
    const float* __restrict__ Emat,     // (12, 4, 4)
    int*         __restrict__ coords,   // (12, 40000) packed iy<<8|ix, -1 = invalid
    float*       __restrict__ mask_out) // (12, 40000)
{
  const int tid  = blockIdx.x * blockDim.x + threadIdx.x;
  const int wave = tid >> 5;
  const int lane = tid & 31;
  const int pl   = lane & 15;
  const int half = lane >> 4;
  const int q    = wave * 16 + pl;       // point index (exact: 2500 waves * 16)

  // B operand: lane<16 -> (K0,K1)=(x,y) ; lane>=16 -> (K2,K3)=(-1,1)
  const float bx = bev[q];
  const float by = bev[KPTS + q];
  v2f b;
  b.x = half ? -1.0f : bx;   // BEV_HEIGHT = -1
  b.y = half ?  1.0f : by;   // homogeneous 1

  const int cam_l = pl >> 2;   // camera within group (0..3)
  const int irow  = pl & 3;    // row of P
  const int j0    = half * 2;  // K index pair held by this lane-half
  const int j1    = j0 + 1;

  for (int grp = 0; grp < 3; ++grp) {
    // A operand: P[i][j] = sum_m I[i][m]*E[m][j] (i<3), P[3][j] = E[3][j]
    const int gA = grp * 4 + cam_l;
    const float* Ib = Imat + gA * 9;
    const float* Eb = Emat + gA * 16;
    const bool top = (irow < 3);
    const int  ii  = top ? irow : 0;                    // keep loads in-bounds
    const float i0 = Ib[ii*3+0], i1 = Ib[ii*3+1], i2 = Ib[ii*3+2];
    v2f a;
    a.x = top ? (i0*Eb[j0] + i1*Eb[4+j0] + i2*Eb[8+j0]) : Eb[12+j0];
    a.y = top ? (i0*Eb[j1] + i1*Eb[4+j1] + i2*Eb[8+j1]) : Eb[12+j1];

    v8f c = {};
    v8f d = __builtin_amdgcn_wmma_f32_16x16x4_f32(
        /*neg_a=*/false, a, /*neg_b=*/false, b,
        /*c_mod=*/(short)0, c, /*reuse_a=*/false, /*reuse_b=*/false);

    // This lane holds D rows M = {half*2, half*2+1 cameras} x 4 rows for point q.
#pragma unroll
    for (int s = 0; s < 2; ++s) {
      const float sp0 = d[s*4 + 0];
      const float sp1 = d[s*4 + 1];
      const float sp2 = d[s*4 + 2];
      const int   gc  = grp*4 + half*2 + s;   // global camera (b*6+n)

      const float z  = sp2;
      const float zz = fmaxf(z, 1e-5f);
      const float xn = (sp0 / zz) / 704.0f;   // / IMAGE_W
      const float yn = (sp1 / zz) / 256.0f;   // / IMAGE_H
      const bool  m  = (z > 1e-5f) && (xn > 0.0f) && (xn < 1.0f)
                                   && (yn > 0.0f) && (yn < 1.0f);

      // ix = round_half_even(((2*xn-1)+1)*w/2 - 0.5), same op order as ref
      const float gx = xn * 2.0f - 1.0f;
      const float gy = yn * 2.0f - 1.0f;
      const float fx = rintf((gx + 1.0f) * 44.0f - 0.5f);  // w/2 = 44
      const float fy = rintf((gy + 1.0f) * 16.0f - 0.5f);  // h/2 = 16
      const bool ok = (fx >= 0.0f) && (fx <= 87.0f) &&
                      (fy >= 0.0f) && (fy <= 31.0f);
      const int packed = ok ? (((int)fy << 8) | (int)fx) : -1;

      coords[gc * KPTS + q]   = packed;
      mask_out[gc * KPTS + q] = m ? 1.0f : 0.0f;
    }
  }
}

// ---------------------------------------------------------------------------
// Stage 1b: NCHW -> NHWC transpose of the feature maps into workspace.
// Only 8.65 MB; writes perfectly coalesced (T[t] linear in thread id),
// reads strided. Makes the per-pixel 64-channel record contiguous (256 B)
// so the big gather can use b128 loads/stores.
// ---------------------------------------------------------------------------
__global__ __launch_bounds__(256) void nchw_to_nhwc(
    const float* __restrict__ images,  // (12, 64, 32, 88)
    float*       __restrict__ T)       // (12, 32*88, 64)
{
  const int t   = blockIdx.x * blockDim.x + threadIdx.x;  // < 12*64*2816
  const int c   = t & 63;
  const int rst = t >> 6;
  const int pix = rst % HW;
  const int g   = rst / HW;
  T[t] = images[(g * IMG_C + c) * HW + pix];
}

// ---------------------------------------------------------------------------
// Stage 2: streaming gather from NHWC, float4 granularity.
// Thread = (camera g, point p, channel-quad ch4).  16 threads (a half-wave)
// cover one point's 64 channels: every tap is a 256 B coalesced b128 load
// from the L2-resident NHWC buffer and a 256 B coalesced non-temporal b128
// store into the 1.1 GB feats stream (write-bandwidth bound @ 23.3 TB/s).
// Clamped-index load + ok-select matches ref's clip + where(ok, feats, 0).
// ---------------------------------------------------------------------------
__global__ __launch_bounds__(256) void bev_gather(
    const float* __restrict__ T,       // (12, 32*88, 64) NHWC
    const int*   __restrict__ coords,  // (12, 40000)
    float*       __restrict__ out)     // (12, 40000, 9, 64)
{
  const long t   = (long)blockIdx.x * blockDim.x + threadIdx.x; // < 12*40000*16
  const int  ch4 = (int)(t & 15);                 // channel quad 0..15
  const long r   = t >> 4;
  const int  p   = (int)(r % KPTS);
  const int  g   = (int)(r / KPTS);

  const int packed = coords[g * KPTS + p];        // half-wave uniform broadcast
  float* o = out + ((long)g * KPTS + p) * 9L * 64L + ch4 * 4;
  const float* img = T + (long)g * HW * 64L;

  if (packed < 0) {
    const v4f zero = {0.0f, 0.0f, 0.0f, 0.0f};
#pragma unroll
    for (int tap = 0; tap < 9; ++tap)
      __builtin_nontemporal_store(zero, (v4f*)(o + (long)tap * 64));
    return;
  }

  const int ix = packed & 0xFF;
  const int iy = packed >> 8;
#pragma unroll
  for (int dy = -1; dy <= 1; ++dy) {
    const int jy = iy + dy;
    const int cy = min(max(jy, 0), IMG_H - 1);
#pragma unroll
    for (int dx = -1; dx <= 1; ++dx) {
      const int jx = ix + dx;
      const int cx = min(max(jx, 0), IMG_W - 1);
      const bool okT = (jy >= 0) && (jy < IMG_H) && (jx >= 0) && (jx < IMG_W);
      v4f v = *(const v4f*)(img + ((long)(cy * IMG_W + cx)) * 64 + ch4 * 4);
      const v4f zero = {0.0f, 0.0f, 0.0f, 0.0f};
      v = okT ? v : zero;
      const int tap = (dy + 1) * 3 + (dx + 1);
      __builtin_nontemporal_store(v, (v4f*)(o + (long)tap * 64));
    }
  }
}

extern "C" void kernel_launch(void* const* d_in, const int* in_sizes, int n_in,
                              void* d_out, int out_size, void* d_ws, size_t ws_size,
                              hipStream_t stream) {
  const float* bev    = (const float*)d_in[0];  // (3,200,200)
  const float* images = (const float*)d_in[1];  // (12,64,32,88)
  const float* Imat   = (const float*)d_in[2];  // (2,6,3,3)
  const float* Emat   = (const float*)d_in[3];  // (2,6,4,4)

  float* out      = (float*)d_out;
  float* mask_out = out + FEATS_ELEMS;          // mask concatenated after feats

  // Workspace layout: [ coords: 12*40000 ints = 1.92 MB | NHWC: 8.65 MB ]
  int*   coords = (int*)d_ws;
  float* T      = (float*)((char*)d_ws + (size_t)NCAM * KPTS * sizeof(int));

  // Stage 1: 2500 waves * 32 lanes = 80000 threads = 625 blocks of 128.
  bev_project_wmma<<<625, 128, 0, stream>>>(bev, Imat, Emat, coords, mask_out);

  // Stage 1b: 12*64*2816 = 2,162,688 elements = 8448 blocks of 256.
  nchw_to_nhwc<<<8448, 256, 0, stream>>>(images, T);

  // Stage 2: 12*40000*16 = 7,680,000 threads = 30000 blocks of 256.
  bev_gather<<<30000, 256, 0, stream>>>(T, coords, out);
}